// SelfAttention_54365696032992
// MI455X (gfx1250) — compile-verified
//
#include <hip/hip_runtime.h>
#include <hip/hip_bf16.h>
#include <math.h>

// ---------------------------------------------------------------------------
// Self-attention with RoPE, fp32 end-to-end on CDNA5 WMMA (v_wmma_f32_16x16x4_f32)
//   B=2, S=2048, C=512, H=32, D=16
// ---------------------------------------------------------------------------

#define B_  2
#define S_  2048
#define C_  512
#define H_  32
#define D_  16
#define HALF_D 8
#define M_TOK (B_ * S_)          // 4096 rows

typedef __attribute__((ext_vector_type(2))) float v2f;
typedef __attribute__((ext_vector_type(8))) float v8f;

static __device__ __forceinline__ v8f wmma4(v2f a, v2f b, v8f c) {
  // 8 args: (neg_a, A, neg_b, B, c_mod, C, reuse_a, reuse_b)
  return __builtin_amdgcn_wmma_f32_16x16x4_f32(false, a, false, b, (short)0, c,
                                               false, false);
}

// -log2(10000)/8  (RoPE: inv_freq[i] = 10000^(-i/8) = exp2(i * RO_LG))
#define RO_LG (-1.6609640474436813f)

// ---------------------------------------------------------------------------
// RoPE trig table: Ctab/Stab[s*8 + i] = cos/sin(s * 10000^(-i/8)).
// One accurate sincosf per (s,i); 128KB total, L2-resident afterwards.
// ---------------------------------------------------------------------------
__global__ void rope_table(float* __restrict__ Ctab, float* __restrict__ Stab) {
  const int idx = blockIdx.x * blockDim.x + threadIdx.x;   // 0 .. S_*8-1
  const int i = idx & (HALF_D - 1);
  const int s = idx >> 3;
  const float ang = (float)s * exp2f((float)i * RO_LG);
  float cs, sn;
  sincosf(ang, &sn, &cs);                 // accurate reduction (angles to ~2047)
  Ctab[idx] = cs;
  Stab[idx] = sn;
}

// ---------------------------------------------------------------------------
// GEMM: Y = X @ W^T,  X is [M_TOK, C_], W is [C_, C_] (row n of W = col n of W^T)
// One wave computes a 16x64 strip: 4 accumulators share one A fragment
// (4 WMMA per 5 global_load_b64).
// mode 0: Y flat [M_TOK, C_]               (final output projection)
// mode 1: RoPE (table lookup) -> [B,H,S,D] (Q, K projections)
// mode 2: write [B,H,S,D], no RoPE         (V projection)
// ---------------------------------------------------------------------------
__global__ void gemm_wmma(const float* __restrict__ X,
                          const float* __restrict__ W,
                          float* __restrict__ Y,
                          const float* __restrict__ Ctab,
                          const float* __restrict__ Stab,
                          int mode) {
  const int lane = threadIdx.x;          // 0..31
  const int half = lane >> 4;            // 0/1
  const int lm   = lane & 15;
  const int m0   = blockIdx.y * 16;      // token-tile
  const int n0   = blockIdx.x * 64;      // 64-wide channel strip

  const float* xrow = X + (size_t)(m0 + lm) * C_ + 2 * half;
  const float* wrow = W + (size_t)(n0 + lm) * C_ + 2 * half;  // +t*16*C_ per tile

  v8f acc[4];
#pragma unroll
  for (int t = 0; t < 4; ++t)
    acc[t] = (v8f){0.f, 0.f, 0.f, 0.f, 0.f, 0.f, 0.f, 0.f};

#pragma unroll 4
  for (int k = 0; k < C_; k += 4) {
    v2f a = *(const v2f*)(xrow + k);     // A frag: M=lm, K=k+2*half+{0,1}
#pragma unroll
    for (int t = 0; t < 4; ++t) {
      v2f b = *(const v2f*)(wrow + (size_t)t * 16 * C_ + k); // B frag: N tile t
      acc[t] = wmma4(a, b, acc[t]);
    }
  }

#pragma unroll
  for (int t = 0; t < 4; ++t) {
    const int n = n0 + t * 16 + lm;      // output channel of this lane
    const int h = n / D_;
    const int d = n & (D_ - 1);
    const int i = d >> 1;                // RoPE frequency index
#pragma unroll
    for (int j = 0; j < 8; ++j) {
      const int m = m0 + j + 8 * half;   // token row (C/D layout)
      float val = acc[t][j];
      if (mode == 0) {
        Y[(size_t)m * C_ + n] = val;
        continue;
      }
      const int b = m / S_;
      const int s = m - b * S_;
      if (mode == 1) {
        // RoPE: pair (2i, 2i+1); adjacent d -> adjacent lanes
        const float cs = Ctab[s * HALF_D + i];
        const float sn = Stab[s * HALF_D + i];
        const float partner = __shfl_xor(val, 1, 32);
        val = (d & 1) ? (val * cs + partner * sn)   // r2 = x2*cos + x1*sin
                      : (val * cs - partner * sn);  // r1 = x1*cos - x2*sin
      }
      // layout [B,H,S,D]
      Y[(((size_t)b * H_ + h) * S_ + s) * D_ + d] = val;
    }
  }
}

// ---------------------------------------------------------------------------
// Causal flash attention per (b,h): one wave handles a 16-row q-tile.
// Qr/Kr/Vw in [B,H,S,D]; O written in [B,S,H,D] (== flat [M_TOK, C_]).
// ---------------------------------------------------------------------------
__global__ void attn_wmma(const float* __restrict__ Qr,
                          const float* __restrict__ Kr,
                          const float* __restrict__ Vw,
                          float* __restrict__ O) {
  __shared__ float plds[16 * 17];        // padded 16x16 P tile

  const int lane = threadIdx.x;
  const int half = lane >> 4;
  const int lm   = lane & 15;
  const int qt   = blockIdx.x;           // q tile index (0..S/16-1)
  const int bh   = blockIdx.y;           // b*H + h
  const int q0   = qt * 16;

  const float* Qb = Qr + (size_t)bh * S_ * D_;
  const float* Kb = Kr + (size_t)bh * S_ * D_;
  const float* Vb = Vw + (size_t)bh * S_ * D_;

  // Q tile as A fragments (resident for whole row of tiles)
  v2f qa[4];
#pragma unroll
  for (int kk = 0; kk < 4; ++kk)
    qa[kk] = *(const v2f*)(Qb + (size_t)(q0 + lm) * D_ + kk * 4 + 2 * half);

  float mrow[8], lrow[8];
#pragma unroll
  for (int j = 0; j < 8; ++j) { mrow[j] = -INFINITY; lrow[j] = 0.f; }
  v8f oacc = {0.f, 0.f, 0.f, 0.f, 0.f, 0.f, 0.f, 0.f};

  const float scale = 0.25f;             // 1/sqrt(D)

  for (int kt = 0; kt <= qt; ++kt) {     // causal: only k-tiles <= q-tile
    const int k0 = kt * 16;

    // K tile as B fragments: element (K=d, N=kcol) = Kr[k0+kcol][d]
    v8f sc = {0.f, 0.f, 0.f, 0.f, 0.f, 0.f, 0.f, 0.f};
#pragma unroll
    for (int kk = 0; kk < 4; ++kk) {
      v2f kb = *(const v2f*)(Kb + (size_t)(k0 + lm) * D_ + kk * 4 + 2 * half);
      sc = wmma4(qa[kk], kb, sc);
    }

    // scale + causal mask (uniform branch: EXEC stays full)
    if (kt == qt) {
#pragma unroll
      for (int j = 0; j < 8; ++j) {
        const int q = q0 + j + 8 * half;
        const int k = k0 + lm;
        sc[j] = (k > q) ? -INFINITY : sc[j] * scale;
      }
    } else {
#pragma unroll
      for (int j = 0; j < 8; ++j) sc[j] *= scale;
    }

    // online softmax, row-wise (rows live in 16-lane halves, xor<=8 stays inside)
#pragma unroll
    for (int j = 0; j < 8; ++j) {
      float sv = sc[j];
      float rm = sv;
      rm = fmaxf(rm, __shfl_xor(rm, 1, 32));
      rm = fmaxf(rm, __shfl_xor(rm, 2, 32));
      rm = fmaxf(rm, __shfl_xor(rm, 4, 32));
      rm = fmaxf(rm, __shfl_xor(rm, 8, 32));
      const float mnew  = fmaxf(mrow[j], rm);
      const float p     = __expf(sv - mnew);
      float rs = p;
      rs += __shfl_xor(rs, 1, 32);
      rs += __shfl_xor(rs, 2, 32);
      rs += __shfl_xor(rs, 4, 32);
      rs += __shfl_xor(rs, 8, 32);
      const float alpha = __expf(mrow[j] - mnew);
      lrow[j] = lrow[j] * alpha + rs;
      mrow[j] = mnew;
      sc[j]   = p;                        // P in C layout
      oacc[j] *= alpha;
    }

    // C layout -> A layout via LDS
#pragma unroll
    for (int j = 0; j < 8; ++j)
      plds[(j + 8 * half) * 17 + lm] = sc[j];
    __syncthreads();

    // P as A fragments, V tile as B fragments; O += P @ V
#pragma unroll
    for (int kk = 0; kk < 4; ++kk) {
      v2f pa;
      pa.x = plds[lm * 17 + kk * 4 + 2 * half];
      pa.y = plds[lm * 17 + kk * 4 + 2 * half + 1];
      v2f vb;
      vb.x = Vb[(size_t)(k0 + kk * 4 + 2 * half) * D_ + lm];
      vb.y = Vb[(size_t)(k0 + kk * 4 + 2 * half + 1) * D_ + lm];
      oacc = wmma4(pa, vb, oacc);
    }
    __syncthreads();
  }

  // normalize + store to [B,S,H,D]
  const int b = bh / H_;
  const int h = bh - b * H_;
#pragma unroll
  for (int j = 0; j < 8; ++j) {
    const int s = q0 + j + 8 * half;
    O[(((size_t)b * S_ + s) * H_ + h) * D_ + lm] = oacc[j] / lrow[j];
  }
}

// ---------------------------------------------------------------------------
extern "C" void kernel_launch(void* const* d_in, const int* in_sizes, int n_in,
                              void* d_out, int out_size, void* d_ws, size_t ws_size,
                              hipStream_t stream) {
  const float* q  = (const float*)d_in[0];
  const float* k  = (const float*)d_in[1];
  const float* v  = (const float*)d_in[2];
  const float* Wq = (const float*)d_in[3];
  const float* Wk = (const float*)d_in[4];
  const float* Wv = (const float*)d_in[5];
  const float* Wo = (const float*)d_in[6];
  float* out = (float*)d_out;

  const size_t BHSD = (size_t)B_ * H_ * S_ * D_;   // 2,097,152 floats
  float* ws = (float*)d_ws;
  float* Qr   = ws;                 // [B,H,S,D]
  float* Kr   = ws + BHSD;          // [B,H,S,D]
  float* Vw   = ws + 2 * BHSD;      // [B,H,S,D]
  float* Ow   = ws + 3 * BHSD;      // [B,S,H,D] == flat [M_TOK, C_]
  float* Ctab = ws + 4 * BHSD;      // [S_, 8]
  float* Stab = Ctab + (size_t)S_ * HALF_D;

  rope_table<<<dim3(S_ * HALF_D / 256), dim3(256), 0, stream>>>(Ctab, Stab);

  dim3 gblk(32);
  dim3 ggrid(C_ / 64, M_TOK / 16);                 // 8 x 256 strips

  gemm_wmma<<<ggrid, gblk, 0, stream>>>(q, Wq, Qr, Ctab, Stab, 1);
  gemm_wmma<<<ggrid, gblk, 0, stream>>>(k, Wk, Kr, Ctab, Stab, 1);
  gemm_wmma<<<ggrid, gblk, 0, stream>>>(v, Wv, Vw, Ctab, Stab, 2);

  dim3 agrid(S_ / 16, B_ * H_);                    // 128 x 64
  attn_wmma<<<agrid, gblk, 0, stream>>>(Qr, Kr, Vw, Ow);

  gemm_wmma<<<ggrid, gblk, 0, stream>>>(Ow, Wo, out, Ctab, Stab, 0);
}